// GraphConvolution_relative_59768764891640
// MI455X (gfx1250) — compile-verified
//
#include <hip/hip_runtime.h>
#include <hip/hip_bf16.h>

// Problem constants (match reference file)
#define NN 8192       // nodes
#define SS 2          // supports
#define EE 262144     // edges per support (2^18)
#define DD 128        // output dim

typedef __attribute__((ext_vector_type(2))) float v2f;
typedef __attribute__((ext_vector_type(8))) float v8f;

// --- Kernel 1: zero the [S, N] diag accumulator in workspace ----------------
__global__ void gc_zero_diag(float* __restrict__ diag) {
    int i = blockIdx.x * blockDim.x + threadIdx.x;
    if (i < SS * NN) diag[i] = 0.0f;
}

// --- Kernel 2: fused gather + segment-sum -----------------------------------
// contrib[s,e] = vals[s,e] * |x[r,c] + x[c,r]|  ; diag[s, r] += contrib
// Never materializes x2 = |x^T + x| (saves ~512 MB of HBM traffic).
__global__ void gc_edge_kernel(const float* __restrict__ x,
                               const int*   __restrict__ rows,
                               const int*   __restrict__ cols,
                               const float* __restrict__ vals,
                               float*       __restrict__ diag) {
    int idx = blockIdx.x * blockDim.x + threadIdx.x;
    if (idx >= SS * EE) return;
    int s = idx >> 18;                 // idx / EE  (EE = 2^18)
    int r = rows[idx];
    int c = cols[idx];
    float v = vals[idx];
    size_t rc = (size_t)r * NN + (size_t)c;
    size_t cr = (size_t)c * NN + (size_t)r;
    float contrib = v * fabsf(x[rc] + x[cr]);
    // hardware global_atomic_add_f32 (order of FP adds is unspecified in the
    // reference segment_sum as well)
    unsafeAtomicAdd(&diag[s * NN + r], contrib);
}

// --- Kernel 3: out[n,d] = relu( sum_s diag[s,n] * w[s,d] ) via WMMA ---------
// [N x 2] @ [2 x 128] GEMM, K padded 2->4 for V_WMMA_F32_16X16X4_F32.
// One wave (32 lanes) computes one 16x16 output tile; 4 waves per block.
__global__ void gc_outer_wmma(const float* __restrict__ diag,
                              const float* __restrict__ w,
                              float*       __restrict__ out) {
    int lane = threadIdx.x;                                   // 0..31
    int n0   = (blockIdx.x * blockDim.y + threadIdx.y) * 16;  // node tile base
    int d0   = blockIdx.y * 16;                               // dim tile base
    int li   = lane & 15;
    bool lo  = lane < 16;

    // A: 16x4 (M x K), lanes 0-15 carry K=0,1 for M=li; lanes 16-31 carry
    // K=2,3 which are zero padding. Loads done by all lanes (safe addresses),
    // lane-dependence via select so EXEC stays all-ones at the WMMA.
    float a0 = diag[n0 + li];          // s = 0
    float a1 = diag[NN + n0 + li];     // s = 1
    // B: 4x16 (K x N), same half-wave split; K=2,3 zero padding.
    float b0 = w[d0 + li];             // s = 0
    float b1 = w[DD + d0 + li];        // s = 1

    v2f a, b;
    a.x = lo ? a0 : 0.0f;  a.y = lo ? a1 : 0.0f;
    b.x = lo ? b0 : 0.0f;  b.y = lo ? b1 : 0.0f;

    v8f c = {};
    // (neg_a, A, neg_b, B, c_mod, C, reuse_a, reuse_b)
    c = __builtin_amdgcn_wmma_f32_16x16x4_f32(
        false, a, false, b, (short)0, c, false, false);

    // C/D layout: VGPR v holds M = v + (lane<16 ? 0 : 8), N = lane&15.
    int mbase = lo ? 0 : 8;
    int col   = d0 + li;
    #pragma unroll
    for (int vv = 0; vv < 8; ++vv) {
        float r = c[vv];
        out[(size_t)(n0 + mbase + vv) * DD + col] = (r > 0.0f) ? r : 0.0f;
    }
}

extern "C" void kernel_launch(void* const* d_in, const int* in_sizes, int n_in,
                              void* d_out, int out_size, void* d_ws, size_t ws_size,
                              hipStream_t stream) {
    const float* x    = (const float*)d_in[0];   // [N, N]
    const int*   rows = (const int*)  d_in[1];   // [S, E]
    const int*   cols = (const int*)  d_in[2];   // [S, E]
    const float* vals = (const float*)d_in[3];   // [S, E]
    const float* w    = (const float*)d_in[4];   // [S, 1, D]
    float* out  = (float*)d_out;                 // [N, D]
    float* diag = (float*)d_ws;                  // [S, N] accumulator (64 KB)

    // 1) zero the accumulator
    gc_zero_diag<<<(SS * NN + 255) / 256, 256, 0, stream>>>(diag);

    // 2) edge gather + FP32-atomic segment sum (S*E = 524288 threads)
    gc_edge_kernel<<<(SS * EE + 255) / 256, 256, 0, stream>>>(x, rows, cols, vals, diag);

    // 3) WMMA outer product + ReLU: grid = (N/64 n-tiles x 4 waves, D/16)
    dim3 grid(NN / 64, DD / 16);
    dim3 block(32, 4);
    gc_outer_wmma<<<grid, block, 0, stream>>>(diag, w, out);
}